// LoOP_19662360281623
// MI455X (gfx1250) — compile-verified
//
#include <hip/hip_runtime.h>
#include <float.h>
#include <math.h>

#define KNN 20
#define DIM 512

typedef __attribute__((ext_vector_type(2))) float v2f;
typedef __attribute__((ext_vector_type(8))) float v8f;

// ---------------------------------------------------------------------------
// Phase 1: per train point n compute ||t_n||^2 and exact ||t_n - X||^2.
// One wave per point; lane loads 4x float4 (fully coalesced 512B/lane-group).
// ---------------------------------------------------------------------------
__global__ void phase1_dist(const float* __restrict__ X,
                            const float* __restrict__ T,
                            float* __restrict__ tn2,
                            float* __restrict__ d0sq, int N) {
  const int lane = threadIdx.x & 31;
  int wave = (blockIdx.x * blockDim.x + threadIdx.x) >> 5;
  int nwaves = (gridDim.x * blockDim.x) >> 5;

  float4 xf[4];
#pragma unroll
  for (int i = 0; i < 4; ++i)
    xf[i] = *(const float4*)(X + (i * 32 + lane) * 4);

  for (int n = wave; n < N; n += nwaves) {
    const float* row = T + (size_t)n * DIM;
    float s2 = 0.f, sd = 0.f;
#pragma unroll
    for (int i = 0; i < 4; ++i) {
      float4 t = *(const float4*)(row + (i * 32 + lane) * 4);
      float4 x = xf[i];
      s2 += t.x * t.x + t.y * t.y + t.z * t.z + t.w * t.w;
      float dx = t.x - x.x, dy = t.y - x.y, dz = t.z - x.z, dw = t.w - x.w;
      sd += dx * dx + dy * dy + dz * dz + dw * dw;
    }
#pragma unroll
    for (int m = 16; m > 0; m >>= 1) {
      s2 += __shfl_xor(s2, m, 32);
      sd += __shfl_xor(sd, m, 32);
    }
    if (lane == 0) { tn2[n] = s2; d0sq[n] = sd; }
  }
}

// ---------------------------------------------------------------------------
// Top-20-min helpers.
// ---------------------------------------------------------------------------
__device__ __forceinline__ void ins20(float* lv, int* li, float v, int idx) {
  if (v >= lv[KNN - 1]) return;
  int j = KNN - 1;
  while (j > 0 && lv[j - 1] > v) { lv[j] = lv[j - 1]; li[j] = li[j - 1]; --j; }
  lv[j] = v; li[j] = idx;
}

// Block-wide: extract the 20 global smallest from each thread's sorted local
// top-20 list. Returns (on thread 0) the sum of the 20 extracted values.
__device__ float block_extract20(float* lv, int* li,
                                 float* sV, int* sI, int* sL,
                                 int tid, int nt, float* outV, int* outI) {
  float sum = 0.f;
  int ptr = 0;
  for (int it = 0; it < KNN; ++it) {
    sV[tid] = (ptr < KNN) ? lv[ptr] : FLT_MAX;
    sI[tid] = (ptr < KNN) ? li[ptr] : -1;
    sL[tid] = tid;
    __syncthreads();
    for (int s = nt >> 1; s > 0; s >>= 1) {
      if (tid < s && sV[tid + s] < sV[tid]) {
        sV[tid] = sV[tid + s]; sI[tid] = sI[tid + s]; sL[tid] = sL[tid + s];
      }
      __syncthreads();
    }
    if (tid == 0) {
      sum += sV[0];
      if (outV) outV[it] = sV[0];
      if (outI) outI[it] = sI[0];
    }
    int win = sL[0];
    __syncthreads();
    if (tid == win) ++ptr;
    __syncthreads();
  }
  return sum;
}

// Stage 1: grid (chunks, rows). Each block emits its chunk's 20 smallest.
__global__ void topk_stage1(const float* __restrict__ vals, int N, int chunks,
                            float* __restrict__ candV, int* __restrict__ candI) {
  const int row = blockIdx.y;
  const int chunk = blockIdx.x;
  const int tid = threadIdx.x;
  const float* base = vals + (size_t)row * N;
  int per = (N + chunks - 1) / chunks;
  int lo = chunk * per;
  int hi = lo + per; if (hi > N) hi = N;

  float lv[KNN]; int li[KNN];
#pragma unroll
  for (int j = 0; j < KNN; ++j) { lv[j] = FLT_MAX; li[j] = -1; }
  for (int i = lo + tid; i < hi; i += blockDim.x) ins20(lv, li, base[i], i);

  __shared__ float sV[128]; __shared__ int sI[128]; __shared__ int sL[128];
  float* outV = candV + ((size_t)row * chunks + chunk) * KNN;
  int* outI = candI + ((size_t)row * chunks + chunk) * KNN;
  block_extract20(lv, li, sV, sI, sL, tid, blockDim.x, outV, outI);
}

// Stage 2 (kNN of X): merge candidates -> 20 nearest indices + pdist_x.
__global__ void topk_stage2_knn(const float* __restrict__ candV,
                                const int* __restrict__ candI, int M,
                                int* __restrict__ nnIdx,
                                float* __restrict__ pdx) {
  const int tid = threadIdx.x;
  float lv[KNN]; int li[KNN];
#pragma unroll
  for (int j = 0; j < KNN; ++j) { lv[j] = FLT_MAX; li[j] = -1; }
  for (int i = tid; i < M; i += blockDim.x) ins20(lv, li, candV[i], candI[i]);
  __shared__ float sV[128]; __shared__ int sI[128]; __shared__ int sL[128];
  float sum = block_extract20(lv, li, sV, sI, sL, tid, blockDim.x, nullptr, nnIdx);
  if (tid == 0) *pdx = sqrtf(sum / (float)KNN);  // Lambda = 1
}

// Stage 2 (per neighbor row): sum of 20 smallest sq -> pdist_nb[row].
__global__ void topk_stage2_rows(const float* __restrict__ candV, int M,
                                 float* __restrict__ pnb) {
  const int tid = threadIdx.x;
  const float* cv = candV + (size_t)blockIdx.x * M;
  float lv[KNN]; int li[KNN];
#pragma unroll
  for (int j = 0; j < KNN; ++j) { lv[j] = FLT_MAX; li[j] = -1; }
  for (int i = tid; i < M; i += blockDim.x) ins20(lv, li, cv[i], i);
  __shared__ float sV[128]; __shared__ int sI[128]; __shared__ int sL[128];
  float sum = block_extract20(lv, li, sV, sI, sL, tid, blockDim.x, nullptr, nullptr);
  if (tid == 0) pnb[blockIdx.x] = sqrtf(sum / (float)KNN);
}

// ---------------------------------------------------------------------------
// Gather the 20 neighbors into a 32x512 padded A-matrix (rows 20..31 zero),
// compute ||neighbor||^2 per row.
// ---------------------------------------------------------------------------
__global__ void gather_nb(const float* __restrict__ T,
                          const int* __restrict__ nnIdx,
                          float* __restrict__ nb, float* __restrict__ nb2) {
  const int r = blockIdx.x;          // 32 blocks
  const int tid = threadIdx.x;       // 128 threads
  __shared__ float red[128];
  float part = 0.f;
  if (r < KNN) {
    const float* src = T + (size_t)nnIdx[r] * DIM;
    for (int i = tid; i < DIM; i += 128) {
      float v = src[i];
      nb[r * DIM + i] = v;
      part += v * v;
    }
  } else {
    for (int i = tid; i < DIM; i += 128) nb[r * DIM + i] = 0.f;
  }
  red[tid] = part;
  __syncthreads();
  for (int s = 64; s > 0; s >>= 1) { if (tid < s) red[tid] += red[tid + s]; __syncthreads(); }
  if (tid == 0) nb2[r] = (r < KNN) ? red[0] : 0.f;
}

// ---------------------------------------------------------------------------
// Phase 2: sq[k][n] = nb2[k] + tn2[n] - 2*dot(nb_k, t_n), clamped at 0.
// 32x512 @ 512x16 per wave-tile via V_WMMA_F32_16X16X4_F32 (fp32 XDL path).
// A (16x4, lane m=lane&15): {A[m][k+2h], A[m][k+2h+1]}  (h = lane>>4)
// B (4x16, lane n=lane&15): {B[2h][n],   B[2h+1][n]}    == train[n0+n][k+2h..]
// C/D 16x16: VGPR v, lane -> M = v + 8h, N = lane&15.
// ---------------------------------------------------------------------------
__global__ void sqmm_wmma(const float* __restrict__ T,
                          const float* __restrict__ nb,
                          const float* __restrict__ nb2,
                          const float* __restrict__ tn2,
                          float* __restrict__ sq, int N) {
  const int lane = threadIdx.x & 31;
  const int half = lane >> 4;
  const int col = lane & 15;
  int wave = (blockIdx.x * blockDim.x + threadIdx.x) >> 5;
  int nwaves = (gridDim.x * blockDim.x) >> 5;
  const int ntiles = (N + 15) >> 4;

  for (int tile = wave; tile < ntiles; tile += nwaves) {
    const int n0 = tile << 4;
    int nc = n0 + col;
    int ncc = nc < N ? nc : N - 1;           // clamp tail reads (store is guarded)
    const float* brow = T + (size_t)ncc * DIM + 2 * half;
    const float* a0p = nb + (size_t)col * DIM + 2 * half;
    const float* a1p = a0p + 16 * DIM;

    v8f acc0 = {};
    v8f acc1 = {};
#pragma unroll 8
    for (int k = 0; k < DIM; k += 4) {
      v2f a0 = *(const v2f*)(a0p + k);
      v2f a1 = *(const v2f*)(a1p + k);
      v2f b = *(const v2f*)(brow + k);
      acc0 = __builtin_amdgcn_wmma_f32_16x16x4_f32(false, a0, false, b,
                                                   (short)0, acc0, false, false);
      acc1 = __builtin_amdgcn_wmma_f32_16x16x4_f32(false, a1, false, b,
                                                   (short)0, acc1, false, false);
    }

    float t2 = tn2[ncc];
    if (nc < N) {
#pragma unroll
      for (int v = 0; v < 8; ++v) {          // rows 0..15
        int M = v + 8 * half;
        float s = nb2[M] + t2 - 2.f * acc0[v];
        sq[(size_t)M * N + nc] = fmaxf(s, 0.f);
      }
#pragma unroll
      for (int v = 0; v < 8; ++v) {          // rows 16..31 (only 16..19 kept)
        int M = 16 + v + 8 * half;
        if (M < KNN) {
          float s = nb2[M] + t2 - 2.f * acc1[v];
          sq[(size_t)M * N + nc] = fmaxf(s, 0.f);
        }
      }
    }
  }
}

// ---------------------------------------------------------------------------
// Final scalar: lof = pdist_x / sum(pdist_nb) * K - 1 ; out = max(erf(lof/sqrt2), 0)
// ---------------------------------------------------------------------------
__global__ void finalize_lof(const float* __restrict__ pdx,
                             const float* __restrict__ pnb,
                             float* __restrict__ out) {
  if (threadIdx.x == 0 && blockIdx.x == 0) {
    float nf = 0.f;
#pragma unroll
    for (int i = 0; i < KNN; ++i) nf += pnb[i];
    float lof = (*pdx) / nf * (float)KNN - 1.f;
    float v = erff(lof / 1.41421356237309515f);   // Z = 1
    out[0] = fmaxf(v, 0.f);
  }
}

extern "C" void kernel_launch(void* const* d_in, const int* in_sizes, int n_in,
                              void* d_out, int out_size, void* d_ws, size_t ws_size,
                              hipStream_t stream) {
  const float* X = (const float*)d_in[0];
  const float* T = (const float*)d_in[1];
  const int N = in_sizes[1] / DIM;            // 100000
  const int CHUNKS = 16;

  // Workspace carving (all offsets >= 8B aligned; ws from hipMalloc is aligned).
  float* tn2   = (float*)d_ws;                     // N
  float* d0sq  = tn2 + N;                          // N
  float* sq    = d0sq + N;                         // KNN * N
  float* nb    = sq + (size_t)KNN * N;             // 32 * 512
  float* nb2   = nb + 32 * DIM;                    // 32 (+pad)
  float* candV = nb2 + 64;                         // KNN*CHUNKS*KNN = 6400
  float* pdx   = candV + KNN * CHUNKS * KNN;       // 1 (+pad)
  float* pnb   = pdx + 64;                         // KNN (+pad)
  int*   candI = (int*)(pnb + 64);                 // 6400
  int*   nnIdx = candI + KNN * CHUNKS * KNN;       // KNN

  // 1) distances to X + per-point squared norms (one pass over 205 MB)
  phase1_dist<<<400, 256, 0, stream>>>(X, T, tn2, d0sq, N);

  // 2) 20 nearest neighbors of X (2-stage top-20-min) + pdist_x
  topk_stage1<<<dim3(CHUNKS, 1), 128, 0, stream>>>(d0sq, N, CHUNKS, candV, candI);
  topk_stage2_knn<<<1, 128, 0, stream>>>(candV, candI, CHUNKS * KNN, nnIdx, pdx);

  // 3) gather neighbors into padded 32x512 A-matrix, row norms
  gather_nb<<<32, 128, 0, stream>>>(T, nnIdx, nb, nb2);

  // 4) 20 x N squared-distance matrix via fp32 WMMA (second pass over 205 MB)
  sqmm_wmma<<<625, 128, 0, stream>>>(T, nb, nb2, tn2, sq, N);

  // 5) per-row sum of 20 smallest sq -> pdist_nb (sq_nn trick: no gather needed)
  topk_stage1<<<dim3(CHUNKS, KNN), 128, 0, stream>>>(sq, N, CHUNKS, candV, candI);
  topk_stage2_rows<<<KNN, 128, 0, stream>>>(candV, CHUNKS * KNN, pnb);

  // 6) scalar LoOP epilogue
  finalize_lof<<<1, 32, 0, stream>>>(pdx, pnb, (float*)d_out);

  (void)n_in; (void)out_size; (void)ws_size;
}